// Cheb2_84954453114994
// MI455X (gfx1250) — compile-verified
//
#include <hip/hip_runtime.h>

#define CIN  128
#define HID  128
#define COUT 64

typedef float v2f __attribute__((ext_vector_type(2)));
typedef float v8f __attribute__((ext_vector_type(8)));

// ---------------- degree / normalization ----------------

__global__ void cheb_deg_kernel(const int* __restrict__ src,
                                const float* __restrict__ w,
                                float* __restrict__ deg, int E) {
    int e = blockIdx.x * blockDim.x + threadIdx.x;
    if (e < E) atomicAdd(&deg[src[e]], w[e]);
}

__global__ void cheb_dinv_kernel(const float* __restrict__ deg,
                                 float* __restrict__ dinv, int N) {
    int i = blockIdx.x * blockDim.x + threadIdx.x;
    if (i < N) {
        float d = deg[i];
        dinv[i] = (d > 0.0f) ? rsqrtf(d) : 0.0f;
    }
}

__global__ void cheb_norm_kernel(const int* __restrict__ src,
                                 const int* __restrict__ dst,
                                 const float* __restrict__ w,
                                 const float* __restrict__ dinv,
                                 float* __restrict__ norm, int E) {
    int e = blockIdx.x * blockDim.x + threadIdx.x;
    if (e < E) norm[e] = -dinv[src[e]] * w[e] * dinv[dst[e]];
}

// ---------------- edge scatter: out[dst] += norm[e] * h[src] ----------------
// One wave (32 lanes) per edge; each lane owns a float4 slice of the 128-wide
// feature row -> fully coalesced 512B gather + 4 global_atomic_add_f32 per lane.

__global__ void cheb_scatter_kernel(const float* __restrict__ h,
                                    const int* __restrict__ src,
                                    const int* __restrict__ dst,
                                    const float* __restrict__ norm,
                                    float* __restrict__ out, int E) {
    int gid  = blockIdx.x * blockDim.x + threadIdx.x;
    int wid  = gid >> 5;           // edge id (uniform per wave)
    int lane = threadIdx.x & 31;
    if (wid >= E) return;
    int   s  = src[wid];
    int   d  = dst[wid];
    float nv = norm[wid];
    float4 v = reinterpret_cast<const float4*>(h + (size_t)s * 128)[lane];
    float* o = out + (size_t)d * 128 + lane * 4;
    atomicAdd(o + 0, nv * v.x);
    atomicAdd(o + 1, nv * v.y);
    atomicAdd(o + 2, nv * v.z);
    atomicAdd(o + 3, nv * v.w);
}

// ---------------- fused Chebyshev GEMM via WMMA ----------------
// out[M x NCOLS] = act( A0 @ W[0] + A1 @ W[1] + bias ),  K = 128.
// One wave per 16x16 output tile; 32 K-steps of 4, two v_wmma_f32_16x16x4_f32
// per step (one per Chebyshev tap), bias pre-loaded into the f32 accumulator.
//
// 32-bit A 16x4 layout : lane l holds row l%16, K = 2*(l/16) .. +1  (float2)
// 32-bit B 4x16 layout : lane l holds col l%16, K = 2*(l/16) .. +1
// 32-bit C/D 16x16     : VGPR v <-> M = v + 8*(l/16), N = l%16

template <int NCOLS, bool RELU>
__global__ __launch_bounds__(256) void cheb_gemm_kernel(
    const float* __restrict__ A0, const float* __restrict__ A1,
    const float* __restrict__ W,     // [2][128][NCOLS] row-major
    const float* __restrict__ bias,  // [NCOLS]
    float* __restrict__ out, int nrows) {
    const int lane = threadIdx.x & 31;
    const int wv   = threadIdx.x >> 5;
    const int ntn  = NCOLS / 16;
    const int tile = blockIdx.x * 8 + wv;              // wave-uniform
    const int total = (nrows / 16) * ntn;
    if (tile >= total) return;                         // whole wave exits: EXEC stays all-1s

    const int  mt    = tile / ntn;
    const int  nt    = tile % ntn;
    const int  col   = lane & 15;                      // = A row index too
    const int  kh    = lane >> 4;                      // 0 or 1 -> K pair select
    const long mbase = (long)mt * 16;
    const int  nbase = nt * 16;

    const float* W0 = W;
    const float* W1 = W + 128 * NCOLS;

    const float bv = bias[nbase + col];
    v8f acc;
#pragma unroll
    for (int i = 0; i < 8; ++i) acc[i] = bv;

    const float* a0p = A0 + (mbase + col) * 128 + kh * 2;
    const float* a1p = A1 + (mbase + col) * 128 + kh * 2;

#pragma unroll 4
    for (int k0 = 0; k0 < 128; k0 += 4) {
        const v2f a0 = *reinterpret_cast<const v2f*>(a0p + k0);
        const v2f a1 = *reinterpret_cast<const v2f*>(a1p + k0);
        const int kr = k0 + kh * 2;
        v2f b0, b1;
        b0.x = W0[(kr + 0) * NCOLS + nbase + col];
        b0.y = W0[(kr + 1) * NCOLS + nbase + col];
        b1.x = W1[(kr + 0) * NCOLS + nbase + col];
        b1.y = W1[(kr + 1) * NCOLS + nbase + col];
        acc = __builtin_amdgcn_wmma_f32_16x16x4_f32(false, a0, false, b0,
                                                    (short)0, acc, false, false);
        acc = __builtin_amdgcn_wmma_f32_16x16x4_f32(false, a1, false, b1,
                                                    (short)0, acc, false, false);
    }

#pragma unroll
    for (int v = 0; v < 8; ++v) {
        const int m = v + kh * 8;
        float r = acc[v];
        if (RELU) r = (r > 0.0f) ? r : 0.0f;
        out[(mbase + m) * NCOLS + nbase + col] = r;
    }
}

// ---------------- launcher ----------------

extern "C" void kernel_launch(void* const* d_in, const int* in_sizes, int n_in,
                              void* d_out, int out_size, void* d_ws, size_t ws_size,
                              hipStream_t stream) {
    const float* x   = (const float*)d_in[0];
    const int*   ei  = (const int*)d_in[1];   // [2][E] int32
    const float* ew  = (const float*)d_in[2];
    const float* W1  = (const float*)d_in[3]; // [2][128][128]
    const float* b1  = (const float*)d_in[4];
    const float* W2  = (const float*)d_in[5]; // [2][128][64]
    const float* b2  = (const float*)d_in[6];

    const int N = in_sizes[0] / CIN;
    const int E = in_sizes[2];
    const int* src = ei;
    const int* dst = ei + E;

    // workspace carve-up (512B aligned)
    char*  ws  = (char*)d_ws;
    size_t off = 0;
    auto carve = [&](size_t bytes) -> float* {
        float* p = (float*)(ws + off);
        off += (bytes + 511) & ~(size_t)511;
        return p;
    };
    float* deg  = carve((size_t)N * 4);
    float* dinv = carve((size_t)N * 4);
    float* nrm  = carve((size_t)E * 4);
    float* px   = carve((size_t)N * CIN * 4);
    float* h    = carve((size_t)N * HID * 4);
    float* ph   = carve((size_t)N * HID * 4);

    const int TB = 256;

    // 1) normalized Laplacian edge weights
    hipMemsetAsync(deg, 0, (size_t)N * 4, stream);
    cheb_deg_kernel<<<(E + TB - 1) / TB, TB, 0, stream>>>(src, ew, deg, E);
    cheb_dinv_kernel<<<(N + TB - 1) / TB, TB, 0, stream>>>(deg, dinv, N);
    cheb_norm_kernel<<<(E + TB - 1) / TB, TB, 0, stream>>>(src, dst, ew, dinv, nrm, E);

    // 2) layer 1: px = L_hat @ x ; h = relu(x@W1[0] + px@W1[1] + b1)
    hipMemsetAsync(px, 0, (size_t)N * CIN * 4, stream);
    {
        long threads = (long)E * 32;
        cheb_scatter_kernel<<<(threads + TB - 1) / TB, TB, 0, stream>>>(x, src, dst, nrm, px, E);
    }
    {
        int tiles = (N / 16) * (HID / 16);
        cheb_gemm_kernel<HID, true><<<(tiles + 7) / 8, TB, 0, stream>>>(x, px, W1, b1, h, N);
    }

    // 3) layer 2: ph = L_hat @ h ; out = h@W2[0] + ph@W2[1] + b2
    hipMemsetAsync(ph, 0, (size_t)N * HID * 4, stream);
    {
        long threads = (long)E * 32;
        cheb_scatter_kernel<<<(threads + TB - 1) / TB, TB, 0, stream>>>(h, src, dst, nrm, ph, E);
    }
    {
        int tiles = (N / 16) * (COUT / 16);
        cheb_gemm_kernel<COUT, false><<<(tiles + 7) / 8, TB, 0, stream>>>(h, ph, W2, b2, (float*)d_out, N);
    }
}